// find_correspondence_27376121544788
// MI455X (gfx1250) — compile-verified
//
#include <hip/hip_runtime.h>
#include <stdint.h>

// exp(BETA*t) = exp2(K_LOG2E * t), BETA = 50
#define K_LOG2E 72.13475204444817f

// Device pass: raw v_exp_f32. Host pass (semantic check only): plain exp2f.
#if defined(__HIP_DEVICE_COMPILE__) && __has_builtin(__builtin_amdgcn_exp2f)
#define EXP2F(x) __builtin_amdgcn_exp2f(x)
#else
#define EXP2F(x) exp2f(x)
#endif

#if __has_builtin(__builtin_amdgcn_global_load_async_to_lds_b128) && \
    __has_builtin(__builtin_amdgcn_s_wait_asynccnt)
#define USE_ASYNC 1
#else
#define USE_ASYNC 0
#endif

// Exact pointee type of the async-load builtin's pointer params
// (clang: "__attribute__((__vector_size__(4 * sizeof(int)))) int *").
typedef int v4i_vs __attribute__((__vector_size__(4 * sizeof(int))));

#define DEPTH 8  // async pipeline depth (8 x 16B in flight per thread)

__device__ __forceinline__ float coordn(int i) {  // linspace(-1,1,64)[i]
    return fmaf((float)i, 2.0f / 63.0f, -1.0f);
}

// Branchless online-softmax accumulate: exactly one v_exp_f32 per element.
__device__ __forceinline__ void upd(float v, float xj, float yi,
                                    float& m, float& se, float& sx, float& sy) {
    float vm = fmaxf(m, v);
    float mn = fminf(m, v);
    float e  = EXP2F(K_LOG2E * (mn - vm));  // <= 1, never overflows
    bool  gt = v > m;
    float d  = gt ? 1.0f : e;   // exp weight of the new element
    float f  = gt ? e : 1.0f;   // rescale of old accumulators
    se = fmaf(se, f, d);
    sx = fmaf(sx, f, d * xj);
    sy = fmaf(sy, f, d * yi);
    m  = vm;
}

// One block per (b, y): 64 output pixels, reduce over s = 0..4095.
// corr element (b, s, y, x) at ((b*4096 + s)*64 + y)*64 + x.
__global__ __launch_bounds__(256) void softargmax_rows(
    const float* __restrict__ corr,
    float* __restrict__ grid_out,   // (b, h, w, 2)
    float* __restrict__ flow_out)   // (b, 2, h, w)
{
    const int tid = threadIdx.x;
    const int b   = blockIdx.x >> 6;
    const int y   = blockIdx.x & 63;
    const int c   = tid & 15;   // x-group: pixels 4c..4c+3
    const int r   = tid >> 4;   // s phase 0..15, s = r + 16*t

    const float* base = corr + (((size_t)b * 4096) * 64 + y) * 64 + (c << 2);

    float m0=-1e30f, m1=-1e30f, m2=-1e30f, m3=-1e30f;
    float se0=0.f, se1=0.f, se2=0.f, se3=0.f;
    float sx0=0.f, sx1=0.f, sx2=0.f, sx3=0.f;
    float sy0=0.f, sy1=0.f, sy2=0.f, sy3=0.f;

    __shared__ float red[256 * 16];

#if USE_ASYNC
    __shared__ float4 abuf[DEPTH][256];
    __shared__ float4 sink[256];  // target for tail dummy issues

    // Prologue: fill the pipeline.
    for (int t = 0; t < DEPTH; ++t) {
        const float* g = base + (size_t)(r + (t << 4)) * 4096;
        __builtin_amdgcn_global_load_async_to_lds_b128(
            (v4i_vs*)g, (v4i_vs*)&abuf[t][tid], 0, 0);
    }
    for (int t = 0; t < 256; ++t) {
        // Always issue exactly one load so s_wait_asynccnt(DEPTH) retires
        // the oldest (step t) load each iteration. Tail issues reload the
        // (L2-hot) last row into a sink buffer.
        int  tin = t + DEPTH;
        int  ts  = tin < 256 ? tin : 255;
        float4* l = tin < 256 ? &abuf[tin & (DEPTH - 1)][tid] : &sink[tid];
        const float* g = base + (size_t)(r + (ts << 4)) * 4096;
        __builtin_amdgcn_global_load_async_to_lds_b128(
            (v4i_vs*)g, (v4i_vs*)l, 0, 0);
        __builtin_amdgcn_s_wait_asynccnt(DEPTH);  // step t's data is in LDS

        float4 v = abuf[t & (DEPTH - 1)][tid];    // own-lane slot: no barrier
        int s = r + (t << 4);
        float xj = coordn(s & 63);   // x_normal[s % 64]
        float yi = coordn(s >> 6);   // y_normal[s / 64]
        upd(v.x, xj, yi, m0, se0, sx0, sy0);
        upd(v.y, xj, yi, m1, se1, sx1, sy1);
        upd(v.z, xj, yi, m2, se2, sx2, sy2);
        upd(v.w, xj, yi, m3, se3, sx3, sy3);
    }
#else
#pragma unroll 4
    for (int t = 0; t < 256; ++t) {
        int s = r + (t << 4);
        const float* p = base + (size_t)s * 4096;
        if (t + 4 < 256)
            __builtin_prefetch(p + (size_t)64 * 4096, 0, 3);  // global_prefetch_b8
        float4 v = *(const float4*)p;
        float xj = coordn(s & 63);
        float yi = coordn(s >> 6);
        upd(v.x, xj, yi, m0, se0, sx0, sy0);
        upd(v.y, xj, yi, m1, se1, sx1, sy1);
        upd(v.z, xj, yi, m2, se2, sx2, sy2);
        upd(v.w, xj, yi, m3, se3, sx3, sy3);
    }
#endif

    // Stash per-thread partials: [tid][pixel*4 + {m, se, sx, sy}]
    {
        float* q = &red[tid * 16];
        q[0]=m0;  q[1]=se0;  q[2]=sx0;  q[3]=sy0;
        q[4]=m1;  q[5]=se1;  q[6]=sx1;  q[7]=sy1;
        q[8]=m2;  q[9]=se2;  q[10]=sx2; q[11]=sy2;
        q[12]=m3; q[13]=se3; q[14]=sx3; q[15]=sy3;
    }
    __syncthreads();

    if (tid < 64) {
        const int x  = tid;
        const int cc = x >> 2;
        const int pp = x & 3;
        float M = -1e30f, SE = 0.f, SX = 0.f, SY = 0.f;
#pragma unroll
        for (int rr = 0; rr < 16; ++rr) {
            const float* s4 = &red[((rr << 4) + cc) * 16 + (pp << 2)];
            float mm = s4[0], ee = s4[1], xx = s4[2], yy = s4[3];
            float nm = fmaxf(M, mm);
            float f1 = EXP2F(K_LOG2E * (M - nm));
            float f2 = EXP2F(K_LOG2E * (mm - nm));
            SE = SE * f1 + ee * f2;
            SX = SX * f1 + xx * f2;
            SY = SY * f1 + yy * f2;
            M = nm;
        }
        float gx = SX / SE;
        float gy = SY / SE;

        size_t pix = ((size_t)b * 64 + y) * 64 + x;
        grid_out[pix * 2 + 0] = gx;
        grid_out[pix * 2 + 1] = gy;

        float fx = gx - coordn(x);
        float fy = gy - coordn(y);
        flow_out[(((size_t)b * 2 + 0) * 64 + y) * 64 + x] = fx;
        flow_out[(((size_t)b * 2 + 1) * 64 + y) * 64 + x] = fy;
    }
}

// smoothness[:,0] = |(F(x+1)-F(x-1)) summed over both flow channels| / 2, etc.
__global__ __launch_bounds__(256) void smoothness_kernel(
    const float* __restrict__ flow, float* __restrict__ sm)
{
    int idx = blockIdx.x * 256 + threadIdx.x;
    if (idx >= 16 * 64 * 64) return;
    int x = idx & 63, y = (idx >> 6) & 63, b = idx >> 12;

    const float* Fx = flow + ((size_t)b * 2 + 0) * 4096;
    const float* Fy = flow + ((size_t)b * 2 + 1) * 4096;

    auto at = [](const float* F, int yy, int xx) -> float {
        return (xx >= 0 && xx < 64 && yy >= 0 && yy < 64) ? F[yy * 64 + xx] : 0.0f;
    };

    float dx = (at(Fx, y, x + 1) - at(Fx, y, x - 1) +
                at(Fy, y, x + 1) - at(Fy, y, x - 1)) * 0.5f;
    float dy = (at(Fx, y + 1, x) - at(Fx, y - 1, x) +
                at(Fy, y + 1, x) - at(Fy, y - 1, x)) * 0.5f;

    sm[((size_t)b * 2 + 0) * 4096 + y * 64 + x] = fabsf(dx);
    sm[((size_t)b * 2 + 1) * 4096 + y * 64 + x] = fabsf(dy);
}

extern "C" void kernel_launch(void* const* d_in, const int* in_sizes, int n_in,
                              void* d_out, int out_size, void* d_ws, size_t ws_size,
                              hipStream_t stream) {
    (void)in_sizes; (void)n_in; (void)out_size; (void)d_ws; (void)ws_size;

    const float* corr = (const float*)d_in[0];
    float* out  = (float*)d_out;
    float* grid = out;                       // 16*64*64*2 = 131072
    float* flow = out + 16 * 64 * 64 * 2;    // 16*2*64*64 = 131072
    float* sm   = flow + 16 * 2 * 64 * 64;   // 131072

    softargmax_rows<<<16 * 64, 256, 0, stream>>>(corr, grid, flow);
    smoothness_kernel<<<(16 * 64 * 64 + 255) / 256, 256, 0, stream>>>(flow, sm);
}